// DirectPredictionGNN_12317966205319
// MI455X (gfx1250) — compile-verified
//
#include <hip/hip_runtime.h>
#include <hip/hip_bf16.h>

// ---------------------------------------------------------------------------
// GCN forward (4-layer GCNConv + BN + ReLU + mean-pool + linear) for gfx1250.
// fp32 throughout; dense GEMMs use V_WMMA_F32_16X16X4_F32 (wave32 WMMA).
// Edge scatter is done with f32 global atomics (feature matrices fit in the
// 192MB L2, so gather + atomic traffic stays on-chip).
// ---------------------------------------------------------------------------

#define NNODES  100000
#define NEDGES  1600000
#define FINDIM  32
#define HDIM    64
#define ODIM    64
#define NLAYERS 4
#define BN_EPS  1e-5f

typedef __attribute__((ext_vector_type(2))) float v2f;
typedef __attribute__((ext_vector_type(8))) float v8f;

// ------------------------------- utility -----------------------------------

__global__ void zero_f32(float* __restrict__ p, int n) {
  int i = blockIdx.x * blockDim.x + threadIdx.x;
  if (i < n) p[i] = 0.0f;
}

// deg[dst] += 1 for every edge
__global__ void degree_kernel(const int* __restrict__ dst, float* __restrict__ deg) {
  int e = blockIdx.x * blockDim.x + threadIdx.x;
  if (e < NEDGES) atomicAdd(&deg[dst[e]], 1.0f);
}

// deg -> (deg+1)^-0.5  (self-loop adds 1)
__global__ void dinv_kernel(float* __restrict__ deg) {
  int n = blockIdx.x * blockDim.x + threadIdx.x;
  if (n < NNODES) deg[n] = rsqrtf(deg[n] + 1.0f);
}

// norm_e[e] = dinv[src] * dinv[dst]
__global__ void norme_kernel(const int* __restrict__ src, const int* __restrict__ dst,
                             const float* __restrict__ dinv, float* __restrict__ norm_e) {
  int e = blockIdx.x * blockDim.x + threadIdx.x;
  if (e < NEDGES) norm_e[e] = dinv[src[e]] * dinv[dst[e]];
}

// ------------------------------ WMMA GEMM ----------------------------------
// out[N x 64] = A[N x K] @ W[K x 64] (+ bias).  Row-major everywhere.
// Block = 256 threads = 8 waves; each wave owns one 16x16 output tile:
// 2 row-strips x 4 col-tiles => block covers 32 rows x 64 cols.
// N = 100000 = 3125 * 32 exactly, so no row guard (EXEC stays all-ones,
// which WMMA requires).
//
// VGPR fragment layouts (ISA 7.12.2, wave32):
//  A 16x4 f32  : lanes 0-15 -> M=lane,     {K=k0,k0+1};  lanes 16-31 -> {k0+2,k0+3}
//  B 4x16 f32  : lanes 0-15 -> N=lane,     {K=k0,k0+1};  lanes 16-31 -> {k0+2,k0+3}
//  C/D 16x16   : VGPR v, lanes 0-15 -> M=v, N=lane; lanes 16-31 -> M=v+8
template <int K>
__global__ __launch_bounds__(256)
void gemm_wmma(const float* __restrict__ A, const float* __restrict__ W,
               const float* __restrict__ bias, float* __restrict__ out) {
  const int lane  = threadIdx.x & 31;
  const int wave  = threadIdx.x >> 5;
  const int m0    = blockIdx.x * 32 + (wave >> 2) * 16;
  const int n0    = (wave & 3) * 16;
  const int lhalf = lane >> 4;   // 0: K pair {0,1}, 1: K pair {2,3}
  const int l15   = lane & 15;

  // Preload all B fragments for this col-tile (K/4 x v2f registers).
  v2f b[K / 4];
#pragma unroll
  for (int kb = 0; kb < K / 4; ++kb) {
    const int kr = kb * 4 + lhalf * 2;
    b[kb].x = W[(kr + 0) * HDIM + n0 + l15];
    b[kb].y = W[(kr + 1) * HDIM + n0 + l15];
  }

  v8f c = {};
  const float* arow = A + (m0 + l15) * K + lhalf * 2;  // 8B aligned
#pragma unroll
  for (int kb = 0; kb < K / 4; ++kb) {
    v2f a = *(const v2f*)(arow + kb * 4);
    c = __builtin_amdgcn_wmma_f32_16x16x4_f32(
        /*neg_a=*/false, a, /*neg_b=*/false, b[kb],
        /*c_mod=*/(short)0, c, /*reuse_a=*/false, /*reuse_b=*/false);
  }

  const float bv = bias ? bias[n0 + l15] : 0.0f;
#pragma unroll
  for (int v = 0; v < 8; ++v) {
    const int row = m0 + v + lhalf * 8;
    out[row * HDIM + n0 + l15] = c[v] + bv;
  }
}

// -------------------------- GCN aggregation --------------------------------

// h[n,f] = xw[n,f] * dinv[n]^2 + conv_b[f]   (self-loop term + bias)
__global__ void init_agg_kernel(const float* __restrict__ xw, const float* __restrict__ dinv,
                                const float* __restrict__ convb, float* __restrict__ h) {
  int i = blockIdx.x * blockDim.x + threadIdx.x;
  if (i < NNODES * HDIM) {
    const int n = i >> 6;
    const int f = i & 63;
    const float dv = dinv[n];
    h[i] = xw[i] * dv * dv + convb[f];
  }
}

// 16 threads per edge; each handles 4 features (float4 gather + 4 f32 atomics).
__global__ __launch_bounds__(256)
void scatter_kernel(const int* __restrict__ src, const int* __restrict__ dst,
                    const float* __restrict__ norm_e, const float* __restrict__ xw,
                    float* __restrict__ h) {
  const int t = blockIdx.x * blockDim.x + threadIdx.x;   // E*16 = 25.6M threads
  const int e = t >> 4;
  if (e >= NEDGES) return;
  const int fq = (t & 15) << 2;
  const float w = norm_e[e];
  const float4 v = *(const float4*)(xw + (long long)src[e] * HDIM + fq);
  float* base = h + (long long)dst[e] * HDIM + fq;
  atomicAdd(base + 0, v.x * w);
  atomicAdd(base + 1, v.y * w);
  atomicAdd(base + 2, v.z * w);
  atomicAdd(base + 3, v.w * w);
}

// -------------------------- BatchNorm + ReLU -------------------------------

// Per-feature sum / sumsq over a 256-row window, reduced in LDS, then atomics.
__global__ __launch_bounds__(256)
void stats_kernel(const float* __restrict__ h, float* __restrict__ stats) {
  __shared__ float s1[256];
  __shared__ float s2[256];
  const int f = threadIdx.x & 63;
  const int r0 = blockIdx.x * 256 + (threadIdx.x >> 6);
  int rend = blockIdx.x * 256 + 256;
  if (rend > NNODES) rend = NNODES;
  float a1 = 0.0f, a2 = 0.0f;
  for (int r = r0; r < rend; r += 4) {
    const float v = h[r * HDIM + f];
    a1 += v;
    a2 += v * v;
  }
  s1[threadIdx.x] = a1;
  s2[threadIdx.x] = a2;
  __syncthreads();
  if (threadIdx.x < 64) {
    const float t1 = s1[threadIdx.x] + s1[threadIdx.x + 64] + s1[threadIdx.x + 128] + s1[threadIdx.x + 192];
    const float t2 = s2[threadIdx.x] + s2[threadIdx.x + 64] + s2[threadIdx.x + 128] + s2[threadIdx.x + 192];
    atomicAdd(&stats[threadIdx.x], t1);
    atomicAdd(&stats[64 + threadIdx.x], t2);
  }
}

// ab[f] = gamma*rstd ; ab[64+f] = beta - gamma*rstd*mu.  Re-zeros stats.
__global__ void finalize_stats_kernel(float* __restrict__ stats,
                                      const float* __restrict__ gamma,
                                      const float* __restrict__ beta,
                                      float* __restrict__ ab) {
  const int f = threadIdx.x;
  if (f < HDIM) {
    const float inv_n = 1.0f / (float)NNODES;
    const float mu  = stats[f] * inv_n;
    const float var = stats[64 + f] * inv_n - mu * mu;
    const float a   = gamma[f] * rsqrtf(var + BN_EPS);
    ab[f]      = a;
    ab[64 + f] = beta[f] - a * mu;
    stats[f]      = 0.0f;   // ready for next layer's accumulation
    stats[64 + f] = 0.0f;
  }
}

__global__ void bn_relu_kernel(float* __restrict__ h, const float* __restrict__ ab) {
  int i = blockIdx.x * blockDim.x + threadIdx.x;
  if (i < NNODES * HDIM) {
    const int f = i & 63;
    h[i] = fmaxf(ab[f] * h[i] + ab[64 + f], 0.0f);
  }
}

// ------------------------------ pooling ------------------------------------

__global__ __launch_bounds__(256)
void pool_kernel(const float* __restrict__ h, float* __restrict__ pooled) {
  __shared__ float s1[256];
  const int f = threadIdx.x & 63;
  const int r0 = blockIdx.x * 256 + (threadIdx.x >> 6);
  int rend = blockIdx.x * 256 + 256;
  if (rend > NNODES) rend = NNODES;
  float a1 = 0.0f;
  for (int r = r0; r < rend; r += 4) a1 += h[r * HDIM + f];
  s1[threadIdx.x] = a1;
  __syncthreads();
  if (threadIdx.x < 64)
    atomicAdd(&pooled[threadIdx.x],
              s1[threadIdx.x] + s1[threadIdx.x + 64] + s1[threadIdx.x + 128] + s1[threadIdx.x + 192]);
}

// out[o] = b_out[o] + sum_k (pooled[k]/N) * W_out[k,o]
__global__ void out_kernel(const float* __restrict__ pooled, const float* __restrict__ Wout,
                           const float* __restrict__ bout, float* __restrict__ out) {
  const int o = threadIdx.x;
  if (o < ODIM) {
    const float inv_n = 1.0f / (float)NNODES;
    float acc = bout[o];
    for (int k = 0; k < HDIM; ++k)
      acc += pooled[k] * inv_n * Wout[k * ODIM + o];
    out[o] = acc;
  }
}

// ------------------------------- launcher ----------------------------------

extern "C" void kernel_launch(void* const* d_in, const int* in_sizes, int n_in,
                              void* d_out, int out_size, void* d_ws, size_t ws_size,
                              hipStream_t stream) {
  (void)in_sizes; (void)n_in; (void)out_size; (void)ws_size;

  const float* x        = (const float*)d_in[0];
  const int*   ei       = (const int*)d_in[1];       // [2,E] int32 per harness convention
  const float* W_embed  = (const float*)d_in[2];
  const float* b_embed  = (const float*)d_in[3];
  const float* conv_W   = (const float*)d_in[4];     // [L,H,H]
  const float* conv_b   = (const float*)d_in[5];     // [L,H]
  const float* bn_gamma = (const float*)d_in[6];     // [L,H]
  const float* bn_beta  = (const float*)d_in[7];     // [L,H]
  const float* W_out    = (const float*)d_in[8];
  const float* b_out    = (const float*)d_in[9];
  float*       out      = (float*)d_out;

  const int* src = ei;
  const int* dst = ei + NEDGES;

  // Workspace layout (floats)
  float* ws     = (float*)d_ws;
  float* dinv   = ws;                     // N
  float* norm_e = dinv + NNODES;          // E
  float* h      = norm_e + NEDGES;        // N*H
  float* xw     = h + NNODES * HDIM;      // N*H
  float* stats  = xw + NNODES * HDIM;     // 128
  float* ab     = stats + 128;            // 128
  float* pooled = ab + 128;               // 64

  const int B = 256;
  const int gN   = (NNODES + B - 1) / B;           // 391
  const int gE   = (NEDGES + B - 1) / B;           // 6250
  const int gNH  = (NNODES * HDIM) / B;            // 25000 (exact)
  const int gSc  = (NEDGES * 16) / B;              // 100000 (exact)
  const int gMM  = NNODES / 32;                    // 3125  (exact)

  // --- degree / normalization (recomputed every call: buffers are scratch) --
  zero_f32<<<gN, B, 0, stream>>>(dinv, NNODES);
  zero_f32<<<2, B, 0, stream>>>(stats, 128 + 128 + 64);   // stats, ab, pooled
  degree_kernel<<<gE, B, 0, stream>>>(dst, dinv);
  dinv_kernel<<<gN, B, 0, stream>>>(dinv);
  norme_kernel<<<gE, B, 0, stream>>>(src, dst, dinv, norm_e);

  // --- node embedding: h = x @ W_embed + b_embed ---------------------------
  gemm_wmma<FINDIM><<<gMM, B, 0, stream>>>(x, W_embed, b_embed, h);

  // --- 4 GCNConv + BN + ReLU layers ---------------------------------------
  for (int i = 0; i < NLAYERS; ++i) {
    gemm_wmma<HDIM><<<gMM, B, 0, stream>>>(h, conv_W + i * HDIM * HDIM, nullptr, xw);
    init_agg_kernel<<<gNH, B, 0, stream>>>(xw, dinv, conv_b + i * HDIM, h);
    scatter_kernel<<<gSc, B, 0, stream>>>(src, dst, norm_e, xw, h);
    stats_kernel<<<gN, B, 0, stream>>>(h, stats);
    finalize_stats_kernel<<<1, 64, 0, stream>>>(stats, bn_gamma + i * HDIM, bn_beta + i * HDIM, ab);
    bn_relu_kernel<<<gNH, B, 0, stream>>>(h, ab);
  }

  // --- mean pool + output projection --------------------------------------
  pool_kernel<<<gN, B, 0, stream>>>(h, pooled);
  out_kernel<<<1, 64, 0, stream>>>(pooled, W_out, b_out, out);
}